// AppearanceReconstruction_67207648248004
// MI455X (gfx1250) — compile-verified
//
#include <hip/hip_runtime.h>
#include <hip/hip_bf16.h>
#include <math.h>

typedef __attribute__((ext_vector_type(2))) float v2f;
typedef __attribute__((ext_vector_type(8))) float v8f;

#define C_DIM   768
#define N_SEQ   1024
#define B_DIM   128
#define H_DIM   1536
#define NBLK    768      // 6144 waves / 8 waves-per-block

// ---------------------------------------------------------------------------
// Kernel B (launched FIRST): r = MLP(LayerNorm(mask_token)). One workgroup.
// Every masked slot's reconstruction equals this single 768-vector, and only
// masked slots survive the mask_ind multiply, so the whole batch MLP collapses.
// ---------------------------------------------------------------------------
__global__ void __launch_bounds__(256) mlp_mask_token_kernel(
    const float* __restrict__ mt,  const float* __restrict__ lnw,
    const float* __restrict__ lnb, const float* __restrict__ W1,
    const float* __restrict__ b1,  const float* __restrict__ W2,
    const float* __restrict__ b2,  float* __restrict__ r_out)
{
    __shared__ float xs[C_DIM];
    __shared__ float hs[H_DIM];
    __shared__ float rs[256];
    __shared__ float rs2[256];

    const int t = threadIdx.x;

    // mean / var over 768
    float s = 0.0f, s2 = 0.0f;
    for (int c = t; c < C_DIM; c += 256) {
        float v = mt[c];
        s += v; s2 += v * v;
    }
    rs[t] = s; rs2[t] = s2;
    __syncthreads();
    for (int o = 128; o > 0; o >>= 1) {
        if (t < o) { rs[t] += rs[t + o]; rs2[t] += rs2[t + o]; }
        __syncthreads();
    }
    const float mu  = rs[0] * (1.0f / (float)C_DIM);
    const float var = rs2[0] * (1.0f / (float)C_DIM) - mu * mu;
    const float inv = rsqrtf(var + 1e-5f);

    for (int c = t; c < C_DIM; c += 256) {
        xs[c] = (mt[c] - mu) * inv * lnw[c] + lnb[c];
    }
    __syncthreads();

    // h = gelu_exact(x @ W1 + b1), W1 is (768, 1536) row-major
    for (int j = t; j < H_DIM; j += 256) {
        float acc = b1[j];
        for (int c = 0; c < C_DIM; ++c) {
            acc = fmaf(xs[c], W1[(size_t)c * H_DIM + j], acc);
        }
        hs[j] = 0.5f * acc * (1.0f + erff(acc * 0.70710678118654752440f));
    }
    __syncthreads();

    // r = h @ W2 + b2, W2 is (1536, 768) row-major
    for (int c = t; c < C_DIM; c += 256) {
        float acc = b2[c];
        for (int j = 0; j < H_DIM; ++j) {
            acc = fmaf(hs[j], W2[(size_t)j * C_DIM + c], acc);
        }
        r_out[c] = acc;
    }
}

// ---------------------------------------------------------------------------
// Kernel A: fused pool + loss. Column sums of tf[b] via V_WMMA_F32_16X16X4_F32
// with an all-ones A matrix (result invariant to any K-row permutation in the
// B operand layout). Each wave: one (batch, 16-channel chunk), then computes
// d^2 against r and block-reduces to one partial per block. Lanes 16-31 hold
// duplicate columns (all D rows identical) -> block sum counts each (b,c)
// twice; compensated in the finalize constant.
// ---------------------------------------------------------------------------
__global__ void __launch_bounds__(256) pool_loss_wmma_kernel(
    const float* __restrict__ tf, const float* __restrict__ r,
    float* __restrict__ partials)
{
    __shared__ float red[256];

    const int t    = threadIdx.x;
    const int lane = t & 31;
    const int wave = blockIdx.x * (blockDim.x >> 5) + (t >> 5);
    const int b    = wave / 48;
    const int c0   = (wave % 48) * 16;
    const int col  = lane & 15;
    const int half = lane >> 4;

    const float* __restrict__ base =
        tf + (size_t)b * (N_SEQ * C_DIM) + (size_t)half * C_DIM + c0 + col;

    const v2f a_ones = {1.0f, 1.0f};     // all-ones A matrix, any lane layout
    v8f acc0 = {}; v8f acc1 = {}; v8f acc2 = {}; v8f acc3 = {};

    for (int n = 0; n < N_SEQ; n += 16) {
        const float* __restrict__ p = base + (size_t)n * C_DIM;
        v2f b0, b1, b2, b3;
        b0[0] = p[0 * C_DIM];  b0[1] = p[ 2 * C_DIM];   // rows n..n+3
        b1[0] = p[4 * C_DIM];  b1[1] = p[ 6 * C_DIM];   // rows n+4..n+7
        b2[0] = p[8 * C_DIM];  b2[1] = p[10 * C_DIM];   // rows n+8..n+11
        b3[0] = p[12 * C_DIM]; b3[1] = p[14 * C_DIM];   // rows n+12..n+15

        acc0 = __builtin_amdgcn_wmma_f32_16x16x4_f32(false, a_ones, false, b0,
                                                     (short)0, acc0, false, false);
        acc1 = __builtin_amdgcn_wmma_f32_16x16x4_f32(false, a_ones, false, b1,
                                                     (short)0, acc1, false, false);
        acc2 = __builtin_amdgcn_wmma_f32_16x16x4_f32(false, a_ones, false, b2,
                                                     (short)0, acc2, false, false);
        acc3 = __builtin_amdgcn_wmma_f32_16x16x4_f32(false, a_ones, false, b3,
                                                     (short)0, acc3, false, false);
    }

    // Every D row equals the column sums; element 0 holds column (lane & 15).
    const float mean = ((acc0[0] + acc1[0]) + (acc2[0] + acc3[0]))
                       * (1.0f / (float)N_SEQ);
    const float d = r[c0 + col] - mean;

    red[t] = d * d;
    __syncthreads();
    for (int o = 128; o > 0; o >>= 1) {
        if (t < o) red[t] += red[t + o];
        __syncthreads();
    }
    if (t == 0) partials[blockIdx.x] = red[0];
}

// ---------------------------------------------------------------------------
// Finalize: deterministic reduction of 768 block partials (no float atomics).
// out = sum(partials) * 0.5 (lane dup) * num_masked(2) / C / (B*2 + 1e-8)
// ---------------------------------------------------------------------------
__global__ void __launch_bounds__(256) finalize_kernel(
    const float* __restrict__ partials, float* __restrict__ out)
{
    __shared__ float red[256];
    const int t = threadIdx.x;
    red[t] = partials[t] + partials[t + 256] + partials[t + 512];
    __syncthreads();
    for (int o = 128; o > 0; o >>= 1) {
        if (t < o) red[t] += red[t + o];
        __syncthreads();
    }
    if (t == 0) {
        out[0] = red[0] * (0.5f * 2.0f) / (float)C_DIM
               / ((float)(B_DIM * 2) + 1e-8f);
    }
}

extern "C" void kernel_launch(void* const* d_in, const int* in_sizes, int n_in,
                              void* d_out, int out_size, void* d_ws, size_t ws_size,
                              hipStream_t stream) {
    (void)in_sizes; (void)n_in; (void)out_size; (void)ws_size;
    // setup_inputs order:
    // 0: appearance_tokens (unused — contributions masked to zero),
    // 1: target_features, 2: noise (unused — mask count is constant),
    // 3: mask_token, 4: ln_w, 5: ln_b, 6: W1, 7: b1, 8: W2, 9: b2
    const float* tf  = (const float*)d_in[1];
    const float* mt  = (const float*)d_in[3];
    const float* lnw = (const float*)d_in[4];
    const float* lnb = (const float*)d_in[5];
    const float* W1  = (const float*)d_in[6];
    const float* b1  = (const float*)d_in[7];
    const float* W2  = (const float*)d_in[8];
    const float* b2  = (const float*)d_in[9];

    float* r        = (float*)d_ws;       // 768 floats
    float* partials = r + C_DIM;          // NBLK = 768 floats
    float* out      = (float*)d_out;

    // Stream order guarantees r is ready before the fused pool+loss kernel.
    hipLaunchKernelGGL(mlp_mask_token_kernel, dim3(1), dim3(256), 0, stream,
                       mt, lnw, lnb, W1, b1, W2, b2, r);
    hipLaunchKernelGGL(pool_loss_wmma_kernel, dim3(NBLK), dim3(256), 0, stream,
                       tf, r, partials);
    hipLaunchKernelGGL(finalize_kernel, dim3(1), dim3(256), 0, stream,
                       partials, out);
}